// Flow_85864986182302
// MI455X (gfx1250) — compile-verified
//
#include <hip/hip_runtime.h>
#include <hip/hip_bf16.h>

// ---------------------------------------------------------------------------
// FlowNet-style correlation cost volume (81 offsets, C=128) + 3x3 conv (337->2)
// Pass 1: WMMA f32 16x16x4 cross-correlation per 16-pixel row tile.
//         D = A(prv 16xK) * B(nxt Kx16); dx displacements are diagonals of D.
//         B_lo covers x0-4..x0+11 (serves output rows M=0..7), B_hi covers
//         x0+4..x0+19 (serves M=8..15).  Out-of-range padding is applied at
//         STORE time (per-lane column validity), so all inner-loop loads are
//         unconditional from clamped addresses -> fully pipelined vmem.
// Pass 2: 3x3x337->2 conv in VALU, float4 vmem + b128 LDS weight reads.
// ---------------------------------------------------------------------------

typedef __attribute__((ext_vector_type(2))) float v2f;
typedef __attribute__((ext_vector_type(8))) float v8f;

#define FH 96
#define FW 96
#define FC 128
#define NOFF 81   // 9x9 offsets

// One wave (32 threads) per (b, y, 16-pixel x tile).
__global__ __launch_bounds__(32) void cost_volume_wmma(
    const float* __restrict__ prv, const float* __restrict__ nxt,
    float* __restrict__ cost) {
  const int lane = threadIdx.x;        // 0..31
  const int li   = lane & 15;
  const int half = lane >> 4;
  const int x0   = blockIdx.x * 16;
  const int y    = blockIdx.y;
  const int b    = blockIdx.z;

  // ---- Preload A: prv[b, y, x0+li, :].  A(16x4) layout: lanes 0-15 hold
  // K=k0+0..1, lanes 16-31 hold K=k0+2..3 (2 consecutive channels per lane).
  const float* aptr =
      prv + (((size_t)b * FH + y) * FW + (x0 + li)) * FC + 2 * half;
  v2f areg[32];
#pragma unroll
  for (int k = 0; k < 32; ++k) areg[k] = *(const v2f*)(aptr + k * 4);

  // B tile pixel coordinates (column N = li).
  const int xb_lo = x0 - 4 + li;               // in [-4, 91]
  const int xb_hi = x0 + 4 + li;               // in [ 4, 99]
  const bool lo_ok = (xb_lo >= 0);
  const bool hi_ok = (xb_hi < FW);
  const bool col_ok = half ? hi_ok : lo_ok;    // this lane's column validity
  const int xlo_c = lo_ok ? xb_lo : 0;         // clamped, always addressable
  const int xhi_c = hi_ok ? xb_hi : (FW - 1);

  for (int dyi = 0; dyi < 9; ++dyi) {
    const int ys = y + dyi - 4;
    const bool row_ok = (ys >= 0) && (ys < FH);
    const int ys_c = row_ok ? ys : (ys < 0 ? 0 : FH - 1);
    const size_t rowbase = ((size_t)b * FH + ys_c) * FW;
    const float* blo = nxt + (rowbase + xlo_c) * FC + 2 * half;
    const float* bhi = nxt + (rowbase + xhi_c) * FC + 2 * half;

    // Prefetch next dy row (CDNA5 global_prefetch_b8) while we crunch this one.
    if (dyi < 8) {
      const int ysn = ys + 1;
      if (ysn >= 0 && ysn < FH)
        __builtin_prefetch(nxt + (((size_t)b * FH + ysn) * FW + (x0 + li)) * FC,
                           0, 3);
    }

    v8f acc_lo = {};
    v8f acc_hi = {};
    // Unconditional, branch-free loads: padding handled at store time.
#pragma unroll
    for (int k = 0; k < 32; ++k) {
      const v2f bl = *(const v2f*)(blo + k * 4);
      const v2f bh = *(const v2f*)(bhi + k * 4);
      // D = A x B + C, full fp32 matrix pipe (16x16x4).
      acc_lo = __builtin_amdgcn_wmma_f32_16x16x4_f32(
          false, areg[k], false, bl, (short)0, acc_lo, false, false);
      acc_hi = __builtin_amdgcn_wmma_f32_16x16x4_f32(
          false, areg[k], false, bh, (short)0, acc_hi, false, false);
    }

    // C/D layout: VGPR v, lanes 0-15 -> (M=v, N=li); lanes 16-31 -> (M=v+8).
    // half==0 lanes consume acc_lo rows M=0..7; half==1 lanes consume acc_hi
    // rows M=8..15.  In both cases dx = li - 4 - v, and the nxt pixel for
    // every stored entry is exactly this lane's own B column -> one validity
    // flag per lane reproduces the reference's zero padding.
    const bool valid = row_ok && col_ok;
    v8f acc = half ? acc_hi : acc_lo;
#pragma unroll
    for (int v = 0; v < 8; ++v) {
      const int dx = li - 4 - v;
      if (dx >= -4 && dx <= 4) {
        const int px = x0 + v + 8 * half;
        float c = valid ? acc[v] * (1.0f / 128.0f) : 0.0f;  // channel mean
        c = (c >= 0.0f) ? c : 0.1f * c;                     // leaky_relu(0.1)
        cost[(((size_t)b * FH + y) * FW + px) * NOFF + dyi * 9 + (dx + 4)] = c;
      }
    }
  }
}

// 3x3 conv over feat = [cost(81) | prv(128) | nxt(128)] -> 2 channels + bias.
// conv_w layout HWIO: w[((ky*3+kx)*337 + cin)*2 + co], staged in LDS (24.3 KB).
__global__ __launch_bounds__(256) void conv3x3_337_2(
    const float* __restrict__ cost, const float* __restrict__ prv,
    const float* __restrict__ nxt, const float* __restrict__ w,
    const float* __restrict__ bias, float* __restrict__ out) {
  __shared__ float sw[3 * 3 * 337 * 2];
  for (int i = threadIdx.x; i < 3 * 3 * 337 * 2; i += blockDim.x) sw[i] = w[i];
  __syncthreads();

  const int idx = blockIdx.x * blockDim.x + threadIdx.x;  // pixel id
  const int total = 16 * FH * FW;
  if (idx >= total) return;
  const int x = idx % FW;
  const int y = (idx / FW) % FH;
  const int b = idx / (FW * FH);

  float acc0 = bias[0];
  float acc1 = bias[1];
#pragma unroll
  for (int ky = 0; ky < 3; ++ky) {
    const int yy = y + ky - 1;
    if (yy < 0 || yy >= FH) continue;
#pragma unroll
    for (int kx = 0; kx < 3; ++kx) {
      const int xx = x + kx - 1;
      if (xx < 0 || xx >= FW) continue;
      const float* swp = sw + (ky * 3 + kx) * 337 * 2;
      const size_t p = ((size_t)b * FH + yy) * FW + xx;

      // 81 cost channels (only 4B-aligned -> scalar b32 loads).
      const float* cvp = cost + p * NOFF;
#pragma unroll 3
      for (int c = 0; c < NOFF; ++c) {
        const float f = cvp[c];
        acc0 = fmaf(f, swp[2 * c + 0], acc0);
        acc1 = fmaf(f, swp[2 * c + 1], acc1);
      }
      // prv / nxt: 512B-aligned bases -> float4 (b128) loads; weights read
      // as b128 from LDS (broadcast, conflict-free).
      const float* pp = prv + p * FC;
      const float* swq = swp + NOFF * 2;
#pragma unroll 4
      for (int c = 0; c < FC; c += 4) {
        const float4 f = *(const float4*)(pp + c);
        const float4 w01 = *(const float4*)(swq + 2 * c);
        const float4 w23 = *(const float4*)(swq + 2 * c + 4);
        acc0 = fmaf(f.x, w01.x, acc0); acc1 = fmaf(f.x, w01.y, acc1);
        acc0 = fmaf(f.y, w01.z, acc0); acc1 = fmaf(f.y, w01.w, acc1);
        acc0 = fmaf(f.z, w23.x, acc0); acc1 = fmaf(f.z, w23.y, acc1);
        acc0 = fmaf(f.w, w23.z, acc0); acc1 = fmaf(f.w, w23.w, acc1);
      }
      const float* np = nxt + p * FC;
      const float* swr = swp + (NOFF + FC) * 2;
#pragma unroll 4
      for (int c = 0; c < FC; c += 4) {
        const float4 f = *(const float4*)(np + c);
        const float4 w01 = *(const float4*)(swr + 2 * c);
        const float4 w23 = *(const float4*)(swr + 2 * c + 4);
        acc0 = fmaf(f.x, w01.x, acc0); acc1 = fmaf(f.x, w01.y, acc1);
        acc0 = fmaf(f.y, w01.z, acc0); acc1 = fmaf(f.y, w01.w, acc1);
        acc0 = fmaf(f.z, w23.x, acc0); acc1 = fmaf(f.z, w23.y, acc1);
        acc0 = fmaf(f.w, w23.z, acc0); acc1 = fmaf(f.w, w23.w, acc1);
      }
    }
  }
  out[(size_t)idx * 2 + 0] = acc0;
  out[(size_t)idx * 2 + 1] = acc1;
}

extern "C" void kernel_launch(void* const* d_in, const int* in_sizes, int n_in,
                              void* d_out, int out_size, void* d_ws,
                              size_t ws_size, hipStream_t stream) {
  const float* prv  = (const float*)d_in[0];   // [16,96,96,128]
  const float* nxt  = (const float*)d_in[1];   // [16,96,96,128]
  const float* cw   = (const float*)d_in[2];   // [3,3,337,2]
  const float* cb   = (const float*)d_in[3];   // [2]
  float* out = (float*)d_out;                  // [16,96,96,2]
  float* cost = (float*)d_ws;                  // needs 16*96*96*81*4 = 47.8 MB

  // Pass 1: 6 x-tiles, 96 rows, 16 batches; one wave per tile.
  dim3 g1(FW / 16, FH, 16);
  cost_volume_wmma<<<g1, 32, 0, stream>>>(prv, nxt, cost);

  // Pass 2: one thread per output pixel.
  const int total = 16 * FH * FW;
  conv3x3_337_2<<<(total + 255) / 256, 256, 0, stream>>>(cost, prv, nxt, cw,
                                                         cb, out);
}